// ShapeletBlock_5506148074029
// MI455X (gfx1250) — compile-verified
//
#include <hip/hip_runtime.h>

// ShapeletBlock fused kernel for MI455X (gfx1250, wave32, WMMA + TDM).
// B=32, T=512, V=16, L=16, S=256, D=64, P=32.
// One workgroup per batch element b; 512 threads = 16 waves (one v per wave).
//   Phase 1: TDM (tensor_load_to_lds) streams W (64KB) into LDS while all
//            threads stage x[b] (32KB, transposed+swizzled) with VALU/DS;
//            candidate slabs prefetched into L2 (global_prefetch_b8).
//   Phase 2: wave w == v: dot = Xp(32x16) @ Cand^T(16x256) via
//            v_wmma_f32_16x16x4_f32; dist = ||x||^2 - 2 dot + ||c||^2
//            (norms fall out of the fragment partials + one shfl_xor(16));
//            min over 32 patches, softmax over 256 shapelets.
//   Phase 3: waves 0..3: out(64x16) = W^T(64x256) @ Q(256x16) + bias via WMMA.

typedef float v2f __attribute__((ext_vector_type(2)));
typedef float v8f __attribute__((ext_vector_type(8)));
typedef unsigned int v4u __attribute__((ext_vector_type(4)));
typedef unsigned int v8u __attribute__((ext_vector_type(8)));

#define B_ 32
#define T_ 512
#define V_ 16
#define L_ 16
#define S_ 256
#define D_ 64
#define P_ 32

#define XROW 520   // 512 + 1 pad float per 64 -> kills 16-stride bank conflicts
#define QROW 257   // 256 + 1 -> bank = (n + s) % 64, conflict-free B-frag reads

__global__ __launch_bounds__(512, 1)
void shapelet_fused_kernel(const float* __restrict__ x,
                           const float* __restrict__ cand,
                           const float* __restrict__ W,
                           const float* __restrict__ bias,
                           float* __restrict__ out)
{
    __shared__ float sW[S_ * D_];       // 16384 floats: W[s][d], row-major (TDM dst)
    __shared__ float sXT[V_ * XROW];    // x[b] transposed to [v][t], swizzled
    __shared__ float sQ[V_ * QROW];     // dist-min, then softmax probs, [v][s]
    __shared__ float sXN[16 * 32];      // per-wave ||x_p||^2 for 32 patches
    __shared__ float sBias[D_];

    const int tid  = threadIdx.x;
    const int lane = tid & 31;
    const int wave = tid >> 5;
    const int b    = blockIdx.x;
    const int n    = lane & 15;   // N index inside a 16x16 tile
    const int hi   = lane >> 4;   // lane half (selects K pair / M half)

    // Warm L2 for this wave's candidate slab (16KB = 32 lanes x 512B stride).
    const float* cv = cand + (size_t)wave * S_ * L_;
    __builtin_prefetch(cv + (size_t)lane * 128, 0, 1);

    // ---------------- Phase 1: TDM-stage W, VALU-stage x[b], bias ----------
    if (wave == 0) {
        // Tensor DMA descriptor (ISA 8.3/8.4): 1-D copy of 16384 x 4B into sW.
        const unsigned long long ga = (unsigned long long)(uintptr_t)W;
        const unsigned lds = (unsigned)(uintptr_t)(void*)sW;  // addr[31:0] = LDS offset
        v4u g0;
        g0.x = 1u;                                   // count=1, user descriptor
        g0.y = lds;                                  // lds_addr
        g0.z = (unsigned)ga;                         // global_addr[31:0]
        g0.w = (unsigned)((ga >> 32) & 0x1FFFFFFu)   // global_addr[56:32]
               | 0x80000000u;                        // type=2 ("image")
        v8u g1;
        g1.s0 = 2u << 16;                            // data_size=4B
        g1.s1 = (unsigned)(S_ * D_) << 16;           // tensor_dim0[15:0]=16384
        g1.s2 = 1u << 16;                            // tensor_dim0[31:16]=0, tensor_dim1=1
        g1.s3 = (unsigned)(S_ * D_) << 16;           // tile_dim0=16384
        g1.s4 = 1u;                                  // tile_dim1=1, tile_dim2=0
        g1.s5 = (unsigned)(S_ * D_);                 // tensor_dim0_stride[31:0]
        g1.s6 = 0u;
        g1.s7 = 0u;
        asm volatile("tensor_load_to_lds %0, %1"
                     :: "s"(g0), "s"(g1) : "memory");
        __builtin_amdgcn_s_wait_tensorcnt(0);
    }
    {
        // thread (v = tid&15, t0 = tid>>4) loads x[b][t0+32k][v]:
        // global index = tid + 512*k  -> perfectly coalesced.
        const float* xb = x + (size_t)b * T_ * V_;
        const int v  = tid & 15;
        const int t0 = tid >> 4;
        #pragma unroll
        for (int k = 0; k < 16; ++k) {
            const int t = t0 + 32 * k;
            sXT[v * XROW + t + (t >> 6)] = xb[(size_t)t * V_ + v];
        }
    }
    if (tid < D_) sBias[tid] = bias[tid];
    __syncthreads();

    // ---------------- Phase 2: distances + min + softmax ----------------
    {
        const int v = wave;

        // Preload A fragments (patch matrix) and ||x_p||^2 partials.
        // A[m][k] = xseg[b,v,p=pt*16+m,l=k] ; frag: lane<16 -> k=4kk+{0,1},
        // lane>=16 -> k=4kk+{2,3} (ISA 7.12.2, 32-bit A 16x4).
        v2f afrag[2][4];
        #pragma unroll
        for (int pt = 0; pt < 2; ++pt) {
            float part = 0.f;
            #pragma unroll
            for (int kk = 0; kk < 4; ++kk) {
                const int t  = (pt * 16 + n) * 16 + 4 * kk + 2 * hi;
                const int t1 = t + 1;
                const float a0 = sXT[v * XROW + t  + (t  >> 6)];
                const float a1 = sXT[v * XROW + t1 + (t1 >> 6)];
                afrag[pt][kk].x = a0;
                afrag[pt][kk].y = a1;
                part += a0 * a0 + a1 * a1;
            }
            // ||x_p||^2 for p = pt*16 + n (combine the two K-halves)
            const float xn = part + __shfl_xor(part, 16, 32);
            sXN[wave * 32 + pt * 16 + n] = xn;   // dup write from hi-half, benign
        }
        // Transpose xnorm into per-accumulator-row registers (loop-invariant).
        float xnr[2][8];
        #pragma unroll
        for (int r = 0; r < 8; ++r) {
            xnr[0][r] = sXN[wave * 32 +      r + 8 * hi];
            xnr[1][r] = sXN[wave * 32 + 16 + r + 8 * hi];
        }

        #pragma unroll 1
        for (int st = 0; st < 16; ++st) {
            const int s0 = st * 16;
            // B fragments: B[k][n] = cand[v][s0+n][k]; per-lane float2 loads,
            // accumulate ||c_s||^2 partial for free.
            v2f bfrag[4];
            float cnp = 0.f;
            #pragma unroll
            for (int kk = 0; kk < 4; ++kk) {
                const float* p = cv + (size_t)(s0 + n) * L_ + 4 * kk + 2 * hi;
                const float b0 = p[0];
                const float b1 = p[1];
                bfrag[kk].x = b0;
                bfrag[kk].y = b1;
                cnp += b0 * b0 + b1 * b1;
            }

            // dot(32x16) = A x B : 2 M-tiles, K=16 via 4 x wmma 16x16x4 f32
            v8f acc0 = {};
            v8f acc1 = {};
            #pragma unroll
            for (int kk = 0; kk < 4; ++kk) {
                acc0 = __builtin_amdgcn_wmma_f32_16x16x4_f32(
                    false, afrag[0][kk], false, bfrag[kk], (short)0, acc0, false, false);
                acc1 = __builtin_amdgcn_wmma_f32_16x16x4_f32(
                    false, afrag[1][kk], false, bfrag[kk], (short)0, acc1, false, false);
            }

            // dist = ||x||^2 - 2 dot + ||c||^2 ; min over all 32 patches.
            // C layout: VGPR r holds M = r + 8*hi, N = lane&15.
            float mn = 3.4e38f;
            #pragma unroll
            for (int r = 0; r < 8; ++r) {
                mn = fminf(mn, xnr[0][r] - 2.0f * acc0[r]);
                mn = fminf(mn, xnr[1][r] - 2.0f * acc1[r]);
            }
            mn = fminf(mn, __shfl_xor(mn, 16, 32));
            const float cn = cnp + __shfl_xor(cnp, 16, 32);
            if (hi == 0)
                sQ[v * QROW + s0 + n] = mn + cn;
        }

        // Softmax over the 256 shapelet distances (8 values per lane).
        float vals[8];
        float vmax = -3.4e38f;
        #pragma unroll
        for (int k = 0; k < 8; ++k) {
            vals[k] = sQ[v * QROW + lane + 32 * k];
            vmax = fmaxf(vmax, vals[k]);
        }
        #pragma unroll
        for (int off = 16; off >= 1; off >>= 1)
            vmax = fmaxf(vmax, __shfl_xor(vmax, off, 32));
        float sum = 0.f;
        #pragma unroll
        for (int k = 0; k < 8; ++k) {
            vals[k] = __expf(vals[k] - vmax);
            sum += vals[k];
        }
        #pragma unroll
        for (int off = 16; off >= 1; off >>= 1)
            sum += __shfl_xor(sum, off, 32);
        const float inv = 1.0f / sum;
        #pragma unroll
        for (int k = 0; k < 8; ++k)
            sQ[v * QROW + lane + 32 * k] = vals[k] * inv;
    }
    __syncthreads();

    // ---------------- Phase 3: out = W^T @ Q + bias ----------------
    // 4 waves, each owns one 16-row d-tile; K = 256 = 64 wmma k-steps.
    if (wave < 4) {
        const int d0 = wave * 16;
        v8f acc = {};
        #pragma unroll 4
        for (int ks = 0; ks < 64; ++ks) {
            const int k0 = 4 * ks + 2 * hi;
            v2f a;  // A[m][k] = W[k][d0+m]
            a.x = sW[(k0    ) * D_ + d0 + n];
            a.y = sW[(k0 + 1) * D_ + d0 + n];
            v2f bq; // B[k][nv] = Q[k][nv] stored as sQ[nv*QROW + k]
            bq.x = sQ[n * QROW + k0];
            bq.y = sQ[n * QROW + k0 + 1];
            acc = __builtin_amdgcn_wmma_f32_16x16x4_f32(
                false, a, false, bq, (short)0, acc, false, false);
        }
        float* ob = out + (size_t)b * D_ * V_;
        #pragma unroll
        for (int r = 0; r < 8; ++r) {
            const int m = r + 8 * hi;
            ob[(size_t)(d0 + m) * V_ + n] = acc[r] + sBias[d0 + m];
        }
    }
}

extern "C" void kernel_launch(void* const* d_in, const int* in_sizes, int n_in,
                              void* d_out, int out_size, void* d_ws, size_t ws_size,
                              hipStream_t stream) {
    (void)in_sizes; (void)n_in; (void)d_ws; (void)ws_size; (void)out_size;
    const float* x    = (const float*)d_in[0];
    const float* cand = (const float*)d_in[1];
    const float* W    = (const float*)d_in[2];
    const float* bias = (const float*)d_in[3];
    float* out        = (float*)d_out;
    shapelet_fused_kernel<<<B_, 512, 0, stream>>>(x, cand, W, bias, out);
}